// EGNN_14929306321391
// MI455X (gfx1250) — compile-verified
//
#include <hip/hip_runtime.h>
#include <hip/hip_bf16.h>

typedef __attribute__((ext_vector_type(2))) float v2f;
typedef __attribute__((ext_vector_type(8))) float v8f;

#define FDIM 64
#define K1P  132   // 129 padded to multiple of 4 (odd multiple of 4 -> conflict-free LDS stride)
#define LDP  68    // 64 padded (odd multiple of 4 stride)
#define MT   32    // edges per block-iteration (two 16-row WMMA tiles)

// Two 16x16 output tiles sharing one B fragment stream: D{0,1} += A{0,1}(16xK) * B(Kx16)
// A row tiles at A and A+16*lda. Per-lane indexing per ISA 32-bit fragment layout:
//   element(v) = M[lane%16][k + 2*(lane/16) + v]
template <int KLEN>
__device__ __forceinline__ void gemm_tile_x2(const float* __restrict__ A, int lda,
                                             const float* __restrict__ B, int ldb,
                                             v8f& c0, v8f& c1) {
    const int lane = threadIdx.x & 31;
    const int hi = lane >> 4, lo = lane & 15;
    const float* a0 = A + lo * lda + 2 * hi;
    const float* a1 = a0 + 16 * lda;
    const float* bp = B + lo * ldb + 2 * hi;
#pragma unroll
    for (int k = 0; k < KLEN; k += 4) {
        v2f b;  b.x  = bp[k]; b.y  = bp[k + 1];
        v2f x0; x0.x = a0[k]; x0.y = a0[k + 1];
        v2f x1; x1.x = a1[k]; x1.y = a1[k + 1];
        c0 = __builtin_amdgcn_wmma_f32_16x16x4_f32(false, x0, false, b, (short)0, c0, false, false);
        c1 = __builtin_amdgcn_wmma_f32_16x16x4_f32(false, x1, false, b, (short)0, c1, false, false);
    }
}

// One A tile against three B tiles (gate weights at B, B+64*ldb, B+128*ldb):
// loads each A fragment once per k-step, issues 3 WMMAs.
template <int KLEN>
__device__ __forceinline__ void gemm_tile_3g(const float* __restrict__ A, int lda,
                                             const float* __restrict__ B, int ldb,
                                             v8f& c0, v8f& c1, v8f& c2) {
    const int lane = threadIdx.x & 31;
    const int hi = lane >> 4, lo = lane & 15;
    const float* ap = A + lo * lda + 2 * hi;
    const float* b0 = B + lo * ldb + 2 * hi;
    const float* b1 = b0 + 64 * ldb;
    const float* b2 = b0 + 128 * ldb;
#pragma unroll
    for (int k = 0; k < KLEN; k += 4) {
        v2f a;  a.x  = ap[k]; a.y  = ap[k + 1];
        v2f y0; y0.x = b0[k]; y0.y = b0[k + 1];
        v2f y1; y1.x = b1[k]; y1.y = b1[k + 1];
        v2f y2; y2.x = b2[k]; y2.y = b2[k + 1];
        c0 = __builtin_amdgcn_wmma_f32_16x16x4_f32(false, a, false, y0, (short)0, c0, false, false);
        c1 = __builtin_amdgcn_wmma_f32_16x16x4_f32(false, a, false, y1, (short)0, c1, false, false);
        c2 = __builtin_amdgcn_wmma_f32_16x16x4_f32(false, a, false, y2, (short)0, c2, false, false);
    }
}

__device__ __forceinline__ float silu(float v) { return v / (1.f + __expf(-v)); }
__device__ __forceinline__ float sigm(float v) { return 1.f / (1.f + __expf(-v)); }

__global__ void zero_ws_kernel(float* __restrict__ p, long n) {
    long i = (long)blockIdx.x * blockDim.x + threadIdx.x;
    if (i < n) p[i] = 0.f;
}

// ---------------- Edge kernel: edge MLP + position head + scatter sums -------------
__global__ __launch_bounds__(128, 1)
void egnn_edge_kernel(const float* __restrict__ h, const float* __restrict__ x,
                      const float* __restrict__ r, const float* __restrict__ eij,
                      const int* __restrict__ src, const int* __restrict__ dst,
                      const float* __restrict__ W1, const float* __restrict__ b1,
                      const float* __restrict__ W2, const float* __restrict__ b2,
                      const float* __restrict__ P1, const float* __restrict__ pb1,
                      const float* __restrict__ P2, const float* __restrict__ pb2,
                      float* __restrict__ sumx, float* __restrict__ cnt,
                      float* __restrict__ summ, long E) {
    __shared__ float sW1[64 * K1P];   // [n][k], k padded 129->132 with zeros
    __shared__ float sW2[64 * LDP];
    __shared__ float sP1[64 * LDP];
    __shared__ float sP2[64];
    __shared__ float sb1[64], sb2[64], spb1[64];
    __shared__ float sInp[MT * K1P];  // [edge][k]  (h_src | h_dst | r | 0 0 0)
    __shared__ float sHid[MT * LDP];
    __shared__ float sM[MT * LDP];
    __shared__ float sWsum[MT];
    __shared__ int   sSrc[MT], sDst[MT];

    const int tid  = threadIdx.x;
    const int wv   = tid >> 5;
    const int lane = tid & 31;
    const int hi   = lane >> 4;
    const int lo   = lane & 15;

    // stage weights once per block (amortized over grid-stride tile loop)
    for (int i = tid; i < 64 * K1P; i += 128) {
        int rr = i / K1P, cc = i % K1P;
        sW1[i] = (cc < 129) ? W1[rr * 129 + cc] : 0.f;
    }
    for (int i = tid; i < 64 * LDP; i += 128) {
        int rr = i / LDP, cc = i % LDP;
        sW2[i] = (cc < 64) ? W2[rr * 64 + cc] : 0.f;
        sP1[i] = (cc < 64) ? P1[rr * 64 + cc] : 0.f;
    }
    if (tid < 64) { sP2[tid] = P2[tid]; sb1[tid] = b1[tid]; sb2[tid] = b2[tid]; spb1[tid] = pb1[tid]; }
    __syncthreads();

    const long  nTiles = (E + MT - 1) / MT;
    const float pb2v   = pb2[0];
    const int   n      = wv * 16 + lo;     // this wave's output column

    for (long tile = blockIdx.x; tile < nTiles; tile += gridDim.x) {
        const long e0 = tile * MT;
        if (tid < MT) {
            long e = e0 + tid;
            sSrc[tid]  = (e < E) ? src[e] : 0;
            sWsum[tid] = 0.f;
        } else if (tid < 2 * MT) {
            long e = e0 + (tid - MT);
            sDst[tid - MT] = (e < E) ? dst[e] : 0;
        }
        __builtin_prefetch(src + e0 + MT, 0, 0);   // global_prefetch for next tile's indices
        __builtin_prefetch(dst + e0 + MT, 0, 0);
        __syncthreads();

        // gather input tile [MT x 132]
        for (int i = tid; i < MT * K1P; i += 128) {
            int  ee = i / K1P, cc = i % K1P;
            long e  = e0 + ee;
            float v = 0.f;
            if (e < E) {
                if (cc < 64)        v = h[(long)sSrc[ee] * 64 + cc];
                else if (cc < 128)  v = h[(long)sDst[ee] * 64 + (cc - 64)];
                else if (cc == 128) v = r[e];
            }
            sInp[i] = v;
        }
        __syncthreads();

        // ----- layer 1: hid = SiLU(inp @ W1^T + b1); this wave owns cols [wv*16, wv*16+16)
        {
            v8f c0 = {0.f,0.f,0.f,0.f,0.f,0.f,0.f,0.f}, c1 = c0;
            gemm_tile_x2<K1P>(sInp, K1P, sW1 + (wv * 16) * K1P, K1P, c0, c1);
            const float bias = sb1[n];
#pragma unroll
            for (int v = 0; v < 8; ++v) {
                int m = v + 8 * hi;
                sHid[m * LDP + n]        = silu(c0[v] + bias);
                sHid[(m + 16) * LDP + n] = silu(c1[v] + bias);
            }
        }
        __syncthreads();

        // ----- layer 2: m = SiLU(hid @ W2^T + b2)
        {
            v8f c0 = {0.f,0.f,0.f,0.f,0.f,0.f,0.f,0.f}, c1 = c0;
            gemm_tile_x2<64>(sHid, LDP, sW2 + (wv * 16) * LDP, LDP, c0, c1);
            const float bias = sb2[n];
#pragma unroll
            for (int v = 0; v < 8; ++v) {
                int m = v + 8 * hi;
                sM[m * LDP + n]        = silu(c0[v] + bias);
                sM[(m + 16) * LDP + n] = silu(c1[v] + bias);
            }
        }
        __syncthreads();

        // ----- position head: w = SiLU(m @ P1^T + pb1) @ P2^T + pb2
        {
            v8f c0 = {0.f,0.f,0.f,0.f,0.f,0.f,0.f,0.f}, c1 = c0;
            gemm_tile_x2<64>(sM, LDP, sP1 + (wv * 16) * LDP, LDP, c0, c1);
            const float bias = spb1[n];
            const float p2v  = sP2[n];
#pragma unroll
            for (int v = 0; v < 8; ++v) {
                int m = v + 8 * hi;
                atomicAdd(&sWsum[m],      silu(c0[v] + bias) * p2v);   // ds_add_f32
                atomicAdd(&sWsum[m + 16], silu(c1[v] + bias) * p2v);
            }
        }
        __syncthreads();

        // scatter x_ij and edge counts (one lane per edge)
        if (tid < MT) {
            long e = e0 + tid;
            if (e < E) {
                float wij = sWsum[tid] + pb2v;
                int   s   = sSrc[tid];
                atomicAdd(&cnt[s], 1.f);
#pragma unroll
                for (int d = 0; d < 3; ++d) {
                    float xv = x[(long)s * 3 + d] + eij[e * 3 + d] * wij;
                    atomicAdd(&sumx[(long)s * 3 + d], xv);
                }
            }
        }
        // scatter m_ij
        for (int i = tid; i < MT * 64; i += 128) {
            int  ee = i >> 6, cc = i & 63;
            long e  = e0 + ee;
            if (e < E) atomicAdd(&summ[(long)sSrc[ee] * 64 + cc], sM[ee * LDP + cc]);
        }
        __syncthreads();
    }
}

// ---------------- Node kernel: scatter-mean finalize + GRU -------------
__global__ __launch_bounds__(128, 1)
void egnn_node_kernel(const float* __restrict__ h, const float* __restrict__ x,
                      const float* __restrict__ Wih, const float* __restrict__ Whh,
                      const float* __restrict__ bih, const float* __restrict__ bhh,
                      const float* __restrict__ sumx, const float* __restrict__ cnt,
                      const float* __restrict__ summ,
                      float* __restrict__ out, int N) {
    __shared__ float sWih[192 * LDP];
    __shared__ float sWhh[192 * LDP];
    __shared__ float sMi[16 * LDP];
    __shared__ float sH[16 * LDP];
    __shared__ float sbi[192], sbh[192];

    const int tid  = threadIdx.x;
    const int wv   = tid >> 5;
    const int lane = tid & 31;
    const int hi   = lane >> 4;
    const int lo   = lane & 15;

    for (int i = tid; i < 192 * LDP; i += 128) {
        int rr = i / LDP, cc = i % LDP;
        sWih[i] = (cc < 64) ? Wih[rr * 64 + cc] : 0.f;
        sWhh[i] = (cc < 64) ? Whh[rr * 64 + cc] : 0.f;
    }
    for (int i = tid; i < 192; i += 128) { sbi[i] = bih[i]; sbh[i] = bhh[i]; }
    __syncthreads();

    const int nTiles = (N + 15) >> 4;
    for (int tile = blockIdx.x; tile < nTiles; tile += gridDim.x) {
        const int n0 = tile << 4;

        if (tid < 16) {                       // position outputs
            int nd = n0 + tid;
            if (nd < N) {
                float c = fmaxf(cnt[nd], 1.f);
#pragma unroll
                for (int d = 0; d < 3; ++d) {
                    float sx = sumx[(long)nd * 3 + d] / c;
                    out[(long)nd * 3 + d]               = sx - floorf(sx);  // x' mod 1
                    out[(long)3 * N + (long)nd * 3 + d] = sx - x[(long)nd * 3 + d];
                }
            }
        }
        for (int i = tid; i < 16 * 64; i += 128) {   // m_i tile and h tile
            int nl = i >> 6, cc = i & 63;
            int nd = n0 + nl;
            float mi = 0.f, hv = 0.f;
            if (nd < N) {
                float c = fmaxf(cnt[nd], 1.f);
                mi = summ[(long)nd * 64 + cc] / c;
                hv = h[(long)nd * 64 + cc];
            }
            sMi[nl * LDP + cc] = mi;
            sH[nl * LDP + cc]  = hv;
        }
        __syncthreads();

        // GRU: gi = m_i @ Wih^T + bih ; gh = h @ Whh^T + bhh ; this wave owns F-cols [wv*16..)
        const int nloc = wv * 16 + lo;
        v8f z8 = {0.f,0.f,0.f,0.f,0.f,0.f,0.f,0.f};
        v8f ci0 = z8, ci1 = z8, ci2 = z8, ch0 = z8, ch1 = z8, ch2 = z8;
        gemm_tile_3g<64>(sMi, LDP, sWih + (wv * 16) * LDP, LDP, ci0, ci1, ci2);
        gemm_tile_3g<64>(sH,  LDP, sWhh + (wv * 16) * LDP, LDP, ch0, ch1, ch2);

#pragma unroll
        for (int v = 0; v < 8; ++v) {
            int m  = v + 8 * hi;
            int nd = n0 + m;
            float ir  = ci0[v] + sbi[nloc];
            float iz  = ci1[v] + sbi[64 + nloc];
            float inn = ci2[v] + sbi[128 + nloc];
            float hr  = ch0[v] + sbh[nloc];
            float hz  = ch1[v] + sbh[64 + nloc];
            float hn  = ch2[v] + sbh[128 + nloc];
            float rg = sigm(ir + hr);
            float zg = sigm(iz + hz);
            float ng = tanhf(inn + rg * hn);
            float hv = sH[m * LDP + nloc];
            float hp = (1.f - zg) * ng + zg * hv;
            if (nd < N) out[(long)6 * N + (long)nd * 64 + nloc] = hp;
        }
        __syncthreads();
    }
}

extern "C" void kernel_launch(void* const* d_in, const int* in_sizes, int n_in,
                              void* d_out, int out_size, void* d_ws, size_t ws_size,
                              hipStream_t stream) {
    const float* h   = (const float*)d_in[0];
    const float* x   = (const float*)d_in[1];
    const float* r   = (const float*)d_in[2];
    const float* eij = (const float*)d_in[3];
    const int*   src = (const int*)d_in[4];
    const int*   dst = (const int*)d_in[5];
    const float* W1  = (const float*)d_in[6];
    const float* b1  = (const float*)d_in[7];
    const float* W2  = (const float*)d_in[8];
    const float* b2  = (const float*)d_in[9];
    const float* P1  = (const float*)d_in[10];
    const float* pb1 = (const float*)d_in[11];
    const float* P2  = (const float*)d_in[12];
    const float* pb2 = (const float*)d_in[13];
    const float* Wih = (const float*)d_in[14];
    const float* Whh = (const float*)d_in[15];
    const float* bih = (const float*)d_in[16];
    const float* bhh = (const float*)d_in[17];

    const int  N = in_sizes[0] / FDIM;   // h is [N, 64]
    const long E = in_sizes[2];          // r_ij is [E]

    // workspace layout: sumx[3N] | cnt[N] | summ[64N]
    float* sumx = (float*)d_ws;
    float* cnt  = sumx + (long)3 * N;
    float* summ = cnt + N;
    const long wsf = (long)68 * N;

    float* out = (float*)d_out;

    zero_ws_kernel<<<(int)((wsf + 255) / 256), 256, 0, stream>>>((float*)d_ws, wsf);

    long nTilesE = (E + MT - 1) / MT;
    int  gridE   = (int)(nTilesE < 4096 ? nTilesE : 4096);
    egnn_edge_kernel<<<gridE, 128, 0, stream>>>(h, x, r, eij, src, dst,
                                                W1, b1, W2, b2, P1, pb1, P2, pb2,
                                                sumx, cnt, summ, E);

    int nTilesN = (N + 15) >> 4;
    int gridN   = nTilesN < 2048 ? nTilesN : 2048;
    egnn_node_kernel<<<gridN, 128, 0, stream>>>(h, x, Wih, Whh, bih, bhh,
                                                sumx, cnt, summ, out, N);
}